// EdgeAggregator_75110388073049
// MI455X (gfx1250) — compile-verified
//
#include <hip/hip_runtime.h>

typedef __attribute__((ext_vector_type(2))) float v2f;
typedef __attribute__((ext_vector_type(4))) float v4f;
typedef __attribute__((ext_vector_type(8))) float v8f;

#define N_NODES 4096
#define N_EDGES 65536
#define BATCH   8
#define UNITS   64

// ---------------------------------------------------------------------------
// Kernel 1: recover edge_target_idx[e] = argmax of one-hot row, computed as
// the exact fp32 matmul  idx = onehot[E,N] x iota[N]  using
// V_WMMA_F32_16X16X4_F32.  One wave handles a tile of 16 edge rows and
// accumulates over K = 4096 nodes in steps of 4 (two accumulators for ILP).
//
// A (16x4 f32) documented layout: lane L: M = L%16, K-half = L/16,
//   VGPR0 = K = 2*Khalf+0, VGPR1 = K = 2*Khalf+1.
// B (4x16 f32): row k broadcast (all 16 columns equal kbase+k), placed with
//   k = 2*(L>=16) + vgpr  (same split pattern as every documented B layout).
// D (16x16 f32) documented layout: lane L, vgpr r -> M = r + 8*(L/16).
// ---------------------------------------------------------------------------
__global__ __launch_bounds__(256) void extract_idx_wmma(
    const float* __restrict__ onehot, int* __restrict__ idx) {
  const int lane  = threadIdx.x & 31;
  const int wave  = threadIdx.x >> 5;
  const int ebase = (blockIdx.x * 8 + wave) * 16;   // 16 edges per wave
  const int m     = lane & 15;
  const int khalf = lane >> 4;                      // 0 or 1

  const float* rowp = onehot + (size_t)(ebase + m) * N_NODES + 2 * khalf;

  v8f acc0 = {0.f, 0.f, 0.f, 0.f, 0.f, 0.f, 0.f, 0.f};
  v8f acc1 = {0.f, 0.f, 0.f, 0.f, 0.f, 0.f, 0.f, 0.f};
  const float kb = (float)(2 * khalf);

  for (int k = 0; k < N_NODES; k += 16) {
    #pragma unroll
    for (int s = 0; s < 4; ++s) {
      // A tile: this lane's pair of one-hot entries at columns
      //   (k + 4*s) + 2*khalf + {0,1}   (8-byte aligned)
      v2f a = *(const v2f*)(rowp + k + 4 * s);
      // B tile: iota values kbase + 2*khalf + {0,1}
      v2f b;
      b.x = kb + (float)(k + 4 * s);
      b.y = b.x + 1.0f;
      if (s & 1)
        acc1 = __builtin_amdgcn_wmma_f32_16x16x4_f32(
            false, a, false, b, (short)0, acc1, false, false);
      else
        acc0 = __builtin_amdgcn_wmma_f32_16x16x4_f32(
            false, a, false, b, (short)0, acc0, false, false);
    }
  }

  // D[m][n] == target index of edge m, identical for every column n.
  // Lane 0 holds edges ebase+0..7 (vgprs 0..7), lane 16 holds ebase+8..15.
  if (m == 0) {
    #pragma unroll
    for (int r = 0; r < 8; ++r) {
      float v = acc0[r] + acc1[r];            // exactly one nonzero term total
      idx[ebase + 8 * khalf + r] = (int)(v + 0.5f);
    }
  }
}

// ---------------------------------------------------------------------------
// Kernel 2: zero the output (harness poisons it; must re-zero every call).
// ---------------------------------------------------------------------------
__global__ __launch_bounds__(256) void zero_out(float* __restrict__ out, int n) {
  int i = blockIdx.x * blockDim.x + threadIdx.x;
  if (i < n) out[i] = 0.0f;
}

// ---------------------------------------------------------------------------
// Kernel 3: scatter-add.  One thread per (b, e, u4-group-of-4): loads float4
// from both edge types, sums in registers, 4x global_atomic_add_f32 into
// out[b, idx[e], u].  Output (8 MB) fits in L2, so atomics stay on-chip.
// ---------------------------------------------------------------------------
__global__ __launch_bounds__(256) void scatter_add(
    const float* __restrict__ msgs, const int* __restrict__ idx,
    float* __restrict__ out) {
  const unsigned long long gid =
      (unsigned long long)blockIdx.x * blockDim.x + threadIdx.x;
  const int b  = (int)(gid >> 20);          // E*16 = 2^20 threads per batch
  const int r  = (int)(gid & 0xFFFFFu);
  const int e  = r >> 4;
  const int u4 = r & 15;

  const size_t tstride = (size_t)N_EDGES * UNITS;               // edge-type stride
  const size_t base    = (size_t)b * 2 * tstride + (size_t)e * UNITS + u4 * 4;

  v4f m0 = *(const v4f*)(msgs + base);
  v4f m1 = *(const v4f*)(msgs + base + tstride);

  const int n = idx[e];
  float* o = out + (size_t)b * (N_NODES * UNITS) + (size_t)n * UNITS + u4 * 4;

  atomicAdd(o + 0, m0.x + m1.x);
  atomicAdd(o + 1, m0.y + m1.y);
  atomicAdd(o + 2, m0.z + m1.z);
  atomicAdd(o + 3, m0.w + m1.w);
}

extern "C" void kernel_launch(void* const* d_in, const int* in_sizes, int n_in,
                              void* d_out, int out_size, void* d_ws, size_t ws_size,
                              hipStream_t stream) {
  const float* msgs   = (const float*)d_in[0];   // [8, 2, 65536, 64] fp32
  const float* onehot = (const float*)d_in[1];   // [65536, 4096] fp32 one-hot
  float*       out    = (float*)d_out;           // [8, 1, 4096, 64] fp32
  int*         idx    = (int*)d_ws;              // 65536 ints of scratch

  // 1) index extraction via fp32 WMMA: 4096 waves, one 16-edge tile each.
  extract_idx_wmma<<<dim3(N_EDGES / 16 / 8), dim3(256), 0, stream>>>(onehot, idx);

  // 2) zero output.
  zero_out<<<dim3((out_size + 255) / 256), dim3(256), 0, stream>>>(out, out_size);

  // 3) scatter-add messages (summed over edge types) onto target nodes.
  const unsigned long long total = (unsigned long long)BATCH * N_EDGES * 16;
  scatter_add<<<dim3((unsigned)(total / 256)), dim3(256), 0, stream>>>(msgs, idx, out);
}